// LoReFTIntervention_41695542509573
// MI455X (gfx1250) — compile-verified
//
#include <hip/hip_runtime.h>

// Problem constants (from the reference): B=8, S=4096, D=3072, RANK=16, P=64
#define BATCH 8
#define SEQ   4096
#define DIM   3072
#define RANK  16
#define POSN  64

typedef __attribute__((ext_vector_type(2))) float v2f;
typedef __attribute__((ext_vector_type(4))) float v4f;
typedef __attribute__((ext_vector_type(8))) float v8f;

// ---------------------------------------------------------------------------
// Kernel 1: full-bandwidth streaming copy h -> out (B128, non-temporal).
// This is the roofline term: ~805 MB @ 23.3 TB/s ~= 35 us.
// ---------------------------------------------------------------------------
__global__ __launch_bounds__(256) void lorft_copy_kernel(
    const v4f* __restrict__ src, v4f* __restrict__ dst, long n4) {
  long stride = (long)gridDim.x * blockDim.x;
  for (long i = (long)blockIdx.x * blockDim.x + threadIdx.x; i < n4; i += stride) {
    v4f v = __builtin_nontemporal_load(&src[i]);
    __builtin_nontemporal_store(v, &dst[i]);
  }
}

// ---------------------------------------------------------------------------
// Kernel 2: diff = h_pos @ (W_w - R_w)^T + W_b   via V_WMMA_F32_16X16X4_F32.
// One wave per 16-row tile of the flattened (b,p) index (32 tiles of 512 rows).
// A fragment (16x4 f32): lane L -> row M = L%16; lanes 0-15 hold K={k,k+1},
// lanes 16-31 hold K={k+2,k+3} across the 2 VGPRs.
// B fragment (4x16 f32): lane L -> col N = L%16; same K split per lane half.
// C/D (16x16 f32): VGPR i, lane L -> row M = i + 8*(L/16), col N = L%16.
// ---------------------------------------------------------------------------
__global__ __launch_bounds__(32) void lorft_diff_kernel(
    const float* __restrict__ h, const float* __restrict__ R_w,
    const float* __restrict__ W_w, const float* __restrict__ W_b,
    const int* __restrict__ positions, float* __restrict__ diff_out) {
  const int tile  = blockIdx.x;      // 0..31
  const int lane  = threadIdx.x;     // 0..31 (wave32)
  const int l15   = lane & 15;
  const int khalf = lane >> 4;       // 0: K pair {0,1}; 1: K pair {2,3}

  // A-matrix row for this lane: flattened row index = tile*16 + (lane%16)
  const int flat = tile * 16 + l15;
  const int b    = flat >> 6;        // / POSN
  const int p    = flat & (POSN - 1);
  const int pos  = positions[p];
  const float* __restrict__ hrow = h + ((size_t)b * SEQ + pos) * DIM;
  // B-matrix column for this lane: rank index = lane%16
  const float* __restrict__ ww = W_w + (size_t)l15 * DIM;
  const float* __restrict__ rw = R_w + (size_t)l15 * DIM;

  v8f c0 = {}, c1 = {};
  // Two independent accumulator chains to hide WMMA latency (768 k-steps).
  for (int k = 0; k < DIM; k += 8) {
    int kk0 = k + 2 * khalf;
    int kk1 = k + 4 + 2 * khalf;
    v2f a0, a1, bf0, bf1;
    a0.x  = hrow[kk0];     a0.y  = hrow[kk0 + 1];
    a1.x  = hrow[kk1];     a1.y  = hrow[kk1 + 1];
    bf0.x = ww[kk0]     - rw[kk0];
    bf0.y = ww[kk0 + 1] - rw[kk0 + 1];
    bf1.x = ww[kk1]     - rw[kk1];
    bf1.y = ww[kk1 + 1] - rw[kk1 + 1];
    c0 = __builtin_amdgcn_wmma_f32_16x16x4_f32(false, a0, false, bf0,
                                               (short)0, c0, false, false);
    c1 = __builtin_amdgcn_wmma_f32_16x16x4_f32(false, a1, false, bf1,
                                               (short)0, c1, false, false);
  }
  v8f c = c0 + c1;

  // Add bias (per rank column N = lane%16) and write diff tile [512 x 16].
  const float bias = W_b[l15];
#pragma unroll
  for (int i = 0; i < 8; ++i) {
    int m = i + 8 * khalf;                  // row within tile
    diff_out[(size_t)(tile * 16 + m) * RANK + l15] = c[i] + bias;
  }
}

// ---------------------------------------------------------------------------
// Kernel 3: out_rows = h_pos + diff @ R_w, fused by seeding the WMMA
// accumulator with the original h tile. Grid: (32 row tiles) x (192 col tiles),
// one wave each; 4 chained V_WMMA_F32_16X16X4_F32 cover K=RANK=16.
// ---------------------------------------------------------------------------
__global__ __launch_bounds__(32) void lorft_apply_kernel(
    const float* __restrict__ h, const float* __restrict__ R_w,
    const int* __restrict__ positions, const float* __restrict__ diff,
    float* __restrict__ out) {
  const int tile  = blockIdx.x;          // row tile 0..31
  const int n0    = blockIdx.y * 16;     // column tile base, 0..3056
  const int lane  = threadIdx.x;
  const int l15   = lane & 15;
  const int khalf = lane >> 4;

  // A fragments: A[M][K] = diff[tile*16 + M][K], M = lane%16, K = rank index.
  const float* __restrict__ arow = diff + (size_t)(tile * 16 + l15) * RANK;
  v2f a[4];
#pragma unroll
  for (int j = 0; j < 4; ++j) {
    int kk = 4 * j + 2 * khalf;
    a[j].x = arow[kk];
    a[j].y = arow[kk + 1];
  }

  // Seed C with the original h tile: row m = i + 8*khalf, col n0 + lane%16.
  v8f c;
#pragma unroll
  for (int i = 0; i < 8; ++i) {
    int m    = i + 8 * khalf;
    int flat = tile * 16 + m;
    int b    = flat >> 6;
    int p    = flat & (POSN - 1);
    int pos  = positions[p];
    c[i] = h[((size_t)b * SEQ + pos) * DIM + n0 + l15];
  }

  // B fragments: B[k][n] = R_w[k][n0 + n], n = lane%16.
#pragma unroll
  for (int j = 0; j < 4; ++j) {
    int kk = 4 * j + 2 * khalf;
    v2f bf;
    bf.x = R_w[(size_t)kk * DIM + n0 + l15];
    bf.y = R_w[(size_t)(kk + 1) * DIM + n0 + l15];
    c = __builtin_amdgcn_wmma_f32_16x16x4_f32(false, a[j], false, bf,
                                              (short)0, c, false, false);
  }

  // Store updated rows (overwrites the bulk copy for these rows).
#pragma unroll
  for (int i = 0; i < 8; ++i) {
    int m    = i + 8 * khalf;
    int flat = tile * 16 + m;
    int b    = flat >> 6;
    int p    = flat & (POSN - 1);
    int pos  = positions[p];
    out[((size_t)b * SEQ + pos) * DIM + n0 + l15] = c[i];
  }
}

// ---------------------------------------------------------------------------
extern "C" void kernel_launch(void* const* d_in, const int* in_sizes, int n_in,
                              void* d_out, int out_size, void* d_ws, size_t ws_size,
                              hipStream_t stream) {
  const float* h         = (const float*)d_in[0];
  const float* R_w       = (const float*)d_in[1];
  const float* W_w       = (const float*)d_in[2];
  const float* W_b       = (const float*)d_in[3];
  const int*   positions = (const int*)d_in[4];
  float*       out       = (float*)d_out;
  float*       diff_ws   = (float*)d_ws;   // 512*16 floats = 32 KB scratch

  // 1) Bulk copy: 8*4096*3072 floats = 25,165,824 float4s.
  const long n4 = (long)BATCH * SEQ * DIM / 4;
  lorft_copy_kernel<<<dim3(16384), dim3(256), 0, stream>>>(
      (const v4f*)h, (v4f*)out, n4);

  // 2) diff = h_pos @ (W_w - R_w)^T + W_b  -> workspace (512 x 16)
  lorft_diff_kernel<<<dim3(32), dim3(32), 0, stream>>>(
      h, R_w, W_w, W_b, positions, diff_ws);

  // 3) rows = h_pos + diff @ R_w  -> overwrite the 512 modified rows in out
  lorft_apply_kernel<<<dim3(32, 192), dim3(32), 0, stream>>>(
      h, R_w, positions, diff_ws, out);
}